// VectorLogic_3805341024635
// MI455X (gfx1250) — compile-verified
//
#include <hip/hip_runtime.h>

typedef float v2f __attribute__((ext_vector_type(2)));
typedef float v8f __attribute__((ext_vector_type(8)));

#define E_DIM 32
#define N_DIM 8
// One tile = 2 batches * 8 vectors = 16 rows of length 32 = 512 contiguous floats.
#define TILE_FLOATS (2 * N_DIM * E_DIM)

__global__ void __launch_bounds__(256)
vector_logic_wmma_kernel(const float* __restrict__ a,
                         const float* __restrict__ truth_raw,
                         float* __restrict__ out,
                         int numTiles, int outCount)
{
    const int lane = threadIdx.x & 31;

    // ---- t = truth_raw / ||truth_raw||, one element per lane (E == 32 == wave32) ----
    float x  = truth_raw[lane];
    float ss = x * x;
    #pragma unroll
    for (int off = 16; off > 0; off >>= 1)
        ss += __shfl_xor(ss, off, 32);
    const float t = x / sqrtf(ss);

    // ---- B fragments (loop-invariant): B[k][n] = t[k] for all n ----
    // 4x16 f32 B fragment: VGPR0 = rows K=0 (lanes 0-15) / K=2 (lanes 16-31),
    //                      VGPR1 = rows K=1 / K=3  (mirrors the 16x4 A layout).
    const int koff = (lane >> 4) << 1; // 0 for lanes 0-15, 2 for lanes 16-31
    v2f bfrag[8];
    #pragma unroll
    for (int s = 0; s < 8; ++s) {
        bfrag[s].x = __shfl(t, 4 * s + koff,     32);
        bfrag[s].y = __shfl(t, 4 * s + koff + 1, 32);
    }

    // A fragment addressing: lane l supplies A[l&15][koff + {0,1}] of the 16x4 slice.
    const int m = lane & 15;
    const int aRowOff = m * E_DIM + koff;

    const int wave   = (int)((blockIdx.x * blockDim.x + threadIdx.x) >> 5);
    const int nWaves = (int)((gridDim.x * blockDim.x) >> 5);

    for (int tile = wave; tile < numTiles; tile += nWaves) {
        const float* base = a + (size_t)tile * TILE_FLOATS;

        if (tile + nWaves < numTiles) // uniform branch: EXEC unchanged
            __builtin_prefetch(base + (size_t)nWaves * TILE_FLOATS, 0, 0);

        v8f c = {}; // fp32 accumulator, 16x16
        #pragma unroll
        for (int s = 0; s < 8; ++s) {
            // K-slice [4s, 4s+4): 8-byte aligned b64 load per lane.
            v2f afrag = *(const v2f*)(base + aRowOff + 4 * s);
            c = __builtin_amdgcn_wmma_f32_16x16x4_f32(
                    /*neg_a=*/false, afrag,
                    /*neg_b=*/false, bfrag[s],
                    /*c_mod=*/(short)0, c,
                    /*reuse_a=*/false, /*reuse_b=*/false);
        }

        // All 16 columns of C are identical. Lane 0-15 reg r = C[r][lane] = dot(row r, t);
        // lanes 16-31 hold rows 8-15. Product over 8 rows = per-batch chained result.
        float p = c[0] * c[1] * c[2] * c[3] * c[4] * c[5] * c[6] * c[7];
        p = fminf(fmaxf(p, 0.0f), 1.0f);

        const int outIdx = tile * 2 + (lane >> 4);
        if (m == 0 && outIdx < outCount)
            out[outIdx] = p;
    }
}

extern "C" void kernel_launch(void* const* d_in, const int* in_sizes, int n_in,
                              void* d_out, int out_size, void* d_ws, size_t ws_size,
                              hipStream_t stream) {
    (void)n_in; (void)d_ws; (void)ws_size;
    const float* a         = (const float*)d_in[0];
    const float* truth_raw = (const float*)d_in[1];
    // d_in[2] (false_raw) provably does not affect the output (t ⟂ f, unit norms).
    float* out = (float*)d_out;

    const int batches  = in_sizes[0] / (N_DIM * E_DIM); // 65536
    const int numTiles = batches / 2;                   // 2 batches per WMMA tile

    const int threads = 256;                            // 8 waves per block
    int blocks = 512;                                   // 4096 waves, grid-stride
    const int minWavesNeeded = (numTiles + 7) / 8;
    if (blocks * 8 > minWavesNeeded * 8) { /* keep 512; memory-bound saturation */ }
    if (blocks < 1) blocks = 1;

    vector_logic_wmma_kernel<<<blocks, threads, 0, stream>>>(
        a, truth_raw, out, numTiles, out_size);
}